// GCNResnet_58789512348392
// MI455X (gfx1250) — compile-verified
//
#include <hip/hip_runtime.h>

// MI455X / gfx1250, wave32. Fused per-node flash-attention + BN + GCN + pool.
// All matmuls via V_WMMA_F32_16X16X4_F32 (f32-exact, D=10 padded to 12).

typedef __attribute__((ext_vector_type(2))) float v2f;
typedef __attribute__((ext_vector_type(8))) float v8f;

#define B_SZ 4096
#define NNODE 3
#define DF 10
#define XSTRIDE (NNODE * DF) /* 30 floats per batch row of x[b,n,d] */

__device__ __forceinline__ v8f wmma4(v2f a, v2f b, v8f c) {
  // D = A(16x4) * B(4x16) + C ; 8 args: (neg_a, A, neg_b, B, c_mod, C, reuse_a, reuse_b)
  return __builtin_amdgcn_wmma_f32_16x16x4_f32(
      false, a, false, b, (short)0, c, false, false);
}

__device__ __forceinline__ v2f mk2(float a, float b) {
  v2f r; r.x = a; r.y = b; return r;
}

// Chunk loader shared by the A (K-tile) and B (Q^T-tile) operands of the
// score WMMA: lane (m,h) holds x[row_base + m][d0 + 2h + {0,1}], 0 for d>=DF.
__device__ __forceinline__ v2f load_qk(const float* __restrict__ xn,
                                       int row_base, int d0, int m, int h) {
  int d = d0 + 2 * h;
  v2f r = mk2(0.f, 0.f);
  if (d < DF) {
    const float* p = xn + (row_base + m) * XSTRIDE + d;
    r.x = p[0];
    r.y = p[1];
  }
  return r;
}

// A operand of O^T = V^T * P^T: lane (m,h),j -> x[kbase + 4c + 2h + j][d = m]
__device__ __forceinline__ v2f load_v(const float* __restrict__ xn,
                                      int kbase, int c, int m, int h) {
  v2f r = mk2(0.f, 0.f);
  if (m < DF) {
    const float* p = xn + (kbase + 4 * c + 2 * h) * XSTRIDE + m;
    r.x = p[0];
    r.y = p[XSTRIDE];
  }
  return r;
}

// B operand of final GEMM: W[d' = d0+2h+j][dout = m], zero-padded.
// Zero rows for d'>=DF also cancel the BN-bias pollution in padded columns.
__device__ __forceinline__ v2f load_w(const float* __restrict__ W,
                                      int d0, int m, int h) {
  int d = d0 + 2 * h;
  v2f r = mk2(0.f, 0.f);
  if (d < DF && m < DF) {
    r.x = W[d * DF + m];
    r.y = W[(d + 1) * DF + m]; // d is even, so d+1 <= 9 whenever d < 10
  }
  return r;
}

__global__ __launch_bounds__(256) void gcn_attn_kernel(
    const float* __restrict__ x, const float* __restrict__ Amat,
    const float* __restrict__ W,
    const float* __restrict__ bn_g, const float* __restrict__ bn_b,
    const float* __restrict__ bn_m, const float* __restrict__ bn_v,
    float* __restrict__ out) {
  const int lane = threadIdx.x & 31;
  const int m = lane & 15;   // 16-wide index (query / dout / d depending on op)
  const int h = lane >> 4;   // half-wave select
  const int wave = threadIdx.x >> 5;
  const int n = blockIdx.y;                       // node channel
  const int qbase = (blockIdx.x * 8 + wave) * 16; // 16-query tile per wave
  const float* xn = x + n * DF;

  // Q^T B-operand chunks, constant across the key loop
  v2f q0 = load_qk(xn, qbase, 0, m, h);
  v2f q1 = load_qk(xn, qbase, 4, m, h);
  v2f q2 = load_qk(xn, qbase, 8, m, h);

  // ---- pass 1: online row max & sum(exp) -------------------------------
  // S^T tile layout: s[r] at lane (m,h) = S[q = m][k = r + 8h]
  float mrun = -3.0e38f, lrun = 0.f;
  for (int kb = 0; kb < B_SZ; kb += 16) {
    v8f s = {};
    s = wmma4(load_qk(xn, kb, 0, m, h), q0, s);
    s = wmma4(load_qk(xn, kb, 4, m, h), q1, s);
    s = wmma4(load_qk(xn, kb, 8, m, h), q2, s);
    float tmax = s[0];
#pragma unroll
    for (int r = 1; r < 8; ++r) tmax = fmaxf(tmax, s[r]);
    float mnew = fmaxf(mrun, tmax);
    float acc = 0.f;
#pragma unroll
    for (int r = 0; r < 8; ++r) acc += __expf(s[r] - mnew);
    lrun = lrun * __expf(mrun - mnew) + acc;
    mrun = mnew;
  }
  // merge the two half-wave partials (each saw half the keys of every tile)
  float mo = __shfl_xor(mrun, 16, 32);
  float mt = fmaxf(mrun, mo);
  float lo = __shfl_xor(lrun, 16, 32);
  float lt = lrun * __expf(mrun - mt) + lo * __expf(mo - mt);
  float invl = 1.0f / lt;

  // ---- pass 2: O^T = sum_k V^T * P^T ; o[r] = v[q=m][d = r+8h] ---------
  v8f o = {};
  for (int kb = 0; kb < B_SZ; kb += 16) {
    v8f s = {};
    s = wmma4(load_qk(xn, kb, 0, m, h), q0, s);
    s = wmma4(load_qk(xn, kb, 4, m, h), q1, s);
    s = wmma4(load_qk(xn, kb, 8, m, h), q2, s);
    float p[8], px[8];
#pragma unroll
    for (int r = 0; r < 8; ++r) p[r] = __expf(s[r] - mt) * invl;
#pragma unroll
    for (int r = 0; r < 8; ++r) px[r] = __shfl_xor(p[r], 16, 32);
    // P^T B-operand chunks: lane (m,h),j -> P[q=m][k = 4c + 2h + j]
    v2f P0 = h ? mk2(px[2], px[3]) : mk2(p[0], p[1]);
    v2f P1 = h ? mk2(px[6], px[7]) : mk2(p[4], p[5]);
    v2f P2 = h ? mk2(p[2], p[3]) : mk2(px[0], px[1]);
    v2f P3 = h ? mk2(p[6], p[7]) : mk2(px[4], px[5]);
    o = wmma4(load_v(xn, kb, 0, m, h), P0, o);
    o = wmma4(load_v(xn, kb, 1, m, h), P1, o);
    o = wmma4(load_v(xn, kb, 2, m, h), P2, o);
    o = wmma4(load_v(xn, kb, 3, m, h), P3, o);
  }

  // ---- eval-mode BatchNorm1d over the node channel ---------------------
  float sg = bn_g[n] * rsqrtf(bn_v[n] + 1e-5f);
  float tb = bn_b[n] - bn_m[n] * sg;
  float ob[8], sx[8];
#pragma unroll
  for (int r = 0; r < 8; ++r) ob[r] = o[r] * sg + tb;
#pragma unroll
  for (int r = 0; r < 8; ++r) sx[r] = __shfl_xor(ob[r], 16, 32);

  // ---- support = vbn @ W via WMMA (A built with the same shuffle trick)
  v2f A0 = h ? mk2(sx[2], sx[3]) : mk2(ob[0], ob[1]);
  v2f A1 = h ? mk2(sx[6], sx[7]) : mk2(ob[4], ob[5]);
  v2f A2 = h ? mk2(ob[2], ob[3]) : mk2(sx[0], sx[1]);
  v8f g = {};
  g = wmma4(A0, load_w(W, 0, m, h), g);
  g = wmma4(A1, load_w(W, 4, m, h), g);
  g = wmma4(A2, load_w(W, 8, m, h), g);

  // adjacency + avg-pool weight: wn = 0.5*(adj[0][n] + adj[1][n]),
  // adj[j][k] = 0.25*((j != k) + A[j][k])
  float a0n = 0.25f * ((n != 0 ? 1.f : 0.f) + Amat[0 * NNODE + n]);
  float a1n = 0.25f * ((n != 1 ? 1.f : 0.f) + Amat[1 * NNODE + n]);
  float wn = 0.5f * (a0n + a1n);

  // g layout: lane (m,h), vgpr r -> contrib[q = qbase + r + 8h][dout = m]
  if (m < DF) {
#pragma unroll
    for (int r = 0; r < 8; ++r) {
      int q = qbase + r + 8 * h;
      atomicAdd(out + q * DF + m, wn * g[r]);
    }
  }
}

extern "C" void kernel_launch(void* const* d_in, const int* in_sizes, int n_in,
                              void* d_out, int out_size, void* d_ws,
                              size_t ws_size, hipStream_t stream) {
  (void)in_sizes; (void)n_in; (void)d_ws; (void)ws_size;
  const float* x  = (const float*)d_in[0];
  const float* A  = (const float*)d_in[1];
  const float* W  = (const float*)d_in[2];
  const float* bg = (const float*)d_in[3];
  const float* bb = (const float*)d_in[4];
  const float* bm = (const float*)d_in[5];
  const float* bv = (const float*)d_in[6];
  float* out = (float*)d_out;

  hipMemsetAsync(out, 0, (size_t)out_size * sizeof(float), stream);
  dim3 grid(B_SZ / (8 * 16), NNODE); // 32 x 3 blocks, 8 waves/block
  gcn_attn_kernel<<<grid, 256, 0, stream>>>(x, A, W, bg, bb, bm, bv, out);
}